// SlotAttention_33809982554115
// MI455X (gfx1250) — compile-verified
//
#include <hip/hip_runtime.h>
#include <hip/hip_bf16.h>

// ---------------------------------------------------------------------------
// Slot Attention, MI455X (gfx1250, wave32).
// Memory-bound problem (>= 64MB input stream per iter vs ~10 GFLOP total), so
// we keep fp32 and use V_WMMA_F32_16X16X4_F32 for the two token-side GEMMs.
// Weight applications are algebraically folded onto the (tiny) slot axis so
// no k/v intermediate (128MB) is ever materialized.
// ---------------------------------------------------------------------------

typedef float v2f __attribute__((ext_vector_type(2)));
typedef float v8f __attribute__((ext_vector_type(8)));

#define B_    64
#define N_    4096
#define D_    64
#define H_    128
#define NS_   8
#define NSP_  16     // slots padded to 16 for WMMA
#define TILE_ 64     // tokens per attn block
#define XSTR_ 68     // LDS row stride (floats): conflict-free column reads
#define ATSTR_ 17    // attn tile stride: conflict-free for all access patterns
#define LN_EPS_ 1e-5f
#define EPS_    1e-8f

// ---------------------------------------------------------------------------
// Per-batch, per-iteration slot-side prep: slots init (iter0), LN, q, and the
// folded projections qk = q*Wk  (so dots = x_ln @ qk^T + qb), qb = q . bk.
// Also zeroes the global accumulators for this iteration.
// ---------------------------------------------------------------------------
__global__ void __launch_bounds__(128)
slot_q_kernel(const float* __restrict__ noise, const float* __restrict__ mu,
              const float* __restrict__ sigma,
              const float* __restrict__ Wq, const float* __restrict__ bq,
              const float* __restrict__ Wk, const float* __restrict__ bk,
              const float* __restrict__ ln_s_w, const float* __restrict__ ln_s_b,
              float* __restrict__ slots_ws, float* __restrict__ qk_ws,
              float* __restrict__ qb_ws, float* __restrict__ raw_ws,
              float* __restrict__ asum_ws, int iter) {
  const int b = blockIdx.x, t = threadIdx.x;
  __shared__ float s[NS_][D_];
  __shared__ float sn[NS_][D_];
  __shared__ float q[NS_][D_];
  __shared__ float mu_s[NS_], rs_s[NS_];

  for (int idx = t; idx < NS_ * D_; idx += 128) {
    const int j = idx >> 6, c = idx & 63;
    float val;
    if (iter == 0) {
      val = mu[c] + sigma[c] * noise[(b * NS_ + j) * D_ + c];
      slots_ws[(b * NS_ + j) * D_ + c] = val;
    } else {
      val = slots_ws[(b * NS_ + j) * D_ + c];
    }
    s[j][c] = val;
  }
  __syncthreads();
  if (t < NS_) {
    float m = 0.f;
    for (int c = 0; c < D_; ++c) m += s[t][c];
    m *= (1.f / D_);
    float v = 0.f;
    for (int c = 0; c < D_; ++c) { float d = s[t][c] - m; v += d * d; }
    v *= (1.f / D_);
    mu_s[t] = m; rs_s[t] = rsqrtf(v + LN_EPS_);
  }
  __syncthreads();
  for (int idx = t; idx < NS_ * D_; idx += 128) {
    const int j = idx >> 6, c = idx & 63;
    sn[j][c] = (s[j][c] - mu_s[j]) * rs_s[j] * ln_s_w[c] + ln_s_b[c];
  }
  __syncthreads();
  for (int idx = t; idx < NS_ * D_; idx += 128) {
    const int j = idx >> 6, c = idx & 63;
    float acc = bq[c];
    for (int cc = 0; cc < D_; ++cc) acc += sn[j][cc] * Wq[c * D_ + cc];
    q[j][c] = acc;
  }
  __syncthreads();
  // qk[j][c2] = sum_c q[j][c] * Wk[c][c2]; rows 8..15 zero (slot padding)
  for (int idx = t; idx < NSP_ * D_; idx += 128) {
    const int j = idx >> 6, c2 = idx & 63;
    float acc = 0.f;
    if (j < NS_)
      for (int c = 0; c < D_; ++c) acc += q[j][c] * Wk[c * D_ + c2];
    qk_ws[(b * NSP_ + j) * D_ + c2] = acc;
  }
  if (t < NSP_) {
    float acc = 0.f;
    if (t < NS_)
      for (int c = 0; c < D_; ++c) acc += q[t][c] * bk[c];
    qb_ws[b * NSP_ + t] = acc;
  }
  for (int idx = t; idx < NS_ * D_; idx += 128) raw_ws[b * NS_ * D_ + idx] = 0.f;
  if (t < NS_) asum_ws[b * NS_ + t] = 0.f;
}

// ---------------------------------------------------------------------------
// Token-side streaming kernel: one 64-token tile per block (4 waves).
//   dots  = x_ln @ qk^T (+qb) * scale   -> WMMA 16x16x4 f32, 4 M-tiles
//   row softmax in LDS (no cross-lane shuffles), +eps, mask
//   raw  += attn^T @ x_ln               -> WMMA 16x16x4 f32, 4 N-tiles
// ---------------------------------------------------------------------------
__global__ void __launch_bounds__(128)
slot_attn_kernel(const float* __restrict__ inputs, const int* __restrict__ mask,
                 const float* __restrict__ ln_in_w, const float* __restrict__ ln_in_b,
                 const float* __restrict__ qk_ws, const float* __restrict__ qb_ws,
                 float* __restrict__ raw_ws, float* __restrict__ asum_ws) {
  const int b = blockIdx.x;
  const int tok0 = blockIdx.y * TILE_;
  const int t = threadIdx.x;
  const int wave = t >> 5;
  const int lane = t & 31;
  const int half = lane >> 4;       // 0: rows g / K lo, 1: rows g+8 / K hi
  const int l16  = lane & 15;

  __shared__ float xs[TILE_][XSTR_];    // LN'd input tile
  __shared__ float qks[NSP_][XSTR_];    // folded slot projection
  __shared__ float qbs[NSP_];
  __shared__ float at[TILE_][ATSTR_];   // dots, then attention (token x slot)
  __shared__ int   mk[TILE_];
  __shared__ float mrow[TILE_], rrow[TILE_];
  __shared__ float ps0[TILE_][2], ps1[TILE_][2];  // LN partial sum / sumsq
  __shared__ float wln[D_], bln[D_];

  const float* inp = inputs + ((size_t)b * N_ + tok0) * D_;
  for (int idx = t; idx < TILE_ * D_ / 4; idx += 128) {
    const int r = idx >> 4, c4 = (idx & 15) << 2;
    const float4 v4 = ((const float4*)(inp + r * D_))[c4 >> 2];
    xs[r][c4 + 0] = v4.x; xs[r][c4 + 1] = v4.y;
    xs[r][c4 + 2] = v4.z; xs[r][c4 + 3] = v4.w;
  }
  for (int idx = t; idx < NSP_ * D_ / 4; idx += 128) {
    const int r = idx >> 4, c4 = (idx & 15) << 2;
    const float4 v4 = ((const float4*)(qk_ws + (b * NSP_ + r) * D_))[c4 >> 2];
    qks[r][c4 + 0] = v4.x; qks[r][c4 + 1] = v4.y;
    qks[r][c4 + 2] = v4.z; qks[r][c4 + 3] = v4.w;
  }
  if (t < NSP_) qbs[t] = qb_ws[b * NSP_ + t];
  if (t < TILE_) mk[t] = mask[(size_t)b * N_ + tok0 + t];
  if (t < D_) { wln[t] = ln_in_w[t]; bln[t] = ln_in_b[t]; }
  // pad columns of the attn tile (slots 8..15) are not zeroed here -- dots stage writes
  // cols 0..15 of all 64 rows (pad columns come out 0)
  __syncthreads();

  // Row LayerNorm stats: 2 threads/row, fused sum + sumsq (one LDS pass)
  {
    const int r = t >> 1, p = t & 1;
    float s = 0.f, sq = 0.f;
    const int c0 = p * 32;
    for (int c = c0; c < c0 + 32; ++c) { const float x = xs[r][c]; s += x; sq += x * x; }
    ps0[r][p] = s; ps1[r][p] = sq;
  }
  __syncthreads();
  if (t < TILE_) {
    const float m = (ps0[t][0] + ps0[t][1]) * (1.f / D_);
    float var = (ps1[t][0] + ps1[t][1]) * (1.f / D_) - m * m;
    var = fmaxf(var, 0.f);
    mrow[t] = m; rrow[t] = rsqrtf(var + LN_EPS_);
  }
  __syncthreads();
  for (int idx = t; idx < TILE_ * D_; idx += 128) {
    const int r = idx >> 6, c = idx & 63;
    xs[r][c] = (xs[r][c] - mrow[r]) * rrow[r] * wln[c] + bln[c];
  }
  __syncthreads();

  // ---- dots = x_ln @ qk^T : wave w -> token M-tile w ----
  {
    const int m0 = wave * 16;
    v8f acc = {};
#pragma unroll
    for (int k0 = 0; k0 < D_; k0 += 4) {
      v2f a, bf;
      a.x = xs[m0 + l16][k0 + 2 * half];
      a.y = xs[m0 + l16][k0 + 2 * half + 1];
      bf.x = qks[l16][k0 + 2 * half];      // B[c,j] = qk[j][c], j = l16
      bf.y = qks[l16][k0 + 2 * half + 1];
      acc = __builtin_amdgcn_wmma_f32_16x16x4_f32(false, a, false, bf,
                                                  (short)0, acc, false, false);
    }
    const float qb_j = qbs[l16];           // zero for pad slots -> dots 0
#pragma unroll
    for (int g = 0; g < 8; ++g)
      at[m0 + g + 8 * half][l16] = (acc[g] + qb_j) * 0.125f;  // scale = d^-0.5
  }
  __syncthreads();

  // ---- per-token softmax over the 8 real slots: 1 thread/row, no shuffles
  if (t < TILE_) {
    float d0[NS_];
    float mx = -3.4e38f;
#pragma unroll
    for (int j = 0; j < NS_; ++j) { d0[j] = at[t][j]; mx = fmaxf(mx, d0[j]); }
    float se = 0.f;
#pragma unroll
    for (int j = 0; j < NS_; ++j) { d0[j] = __expf(d0[j] - mx); se += d0[j]; }
    const float inv = 1.f / se;
    const bool keep = (mk[t] != 0);
#pragma unroll
    for (int j = 0; j < NS_; ++j)
      at[t][j] = keep ? (d0[j] * inv + EPS_) : 0.f;
    // pad slot columns stay exactly zero (written as 0 by the dots stage and
    // untouched here) so the u1 GEMM's padded A-rows contribute nothing
  }
  __syncthreads();

  // ---- raw += attn^T @ x_ln : wave w -> channel N-tile w ----
  {
    const int n0 = wave * 16;
    v8f acc = {};
#pragma unroll
    for (int k0 = 0; k0 < TILE_; k0 += 4) {
      v2f a, bf;
      a.x = at[k0 + 2 * half][l16];              // A[j,tok] = attn[tok][j]
      a.y = at[k0 + 2 * half + 1][l16];
      bf.x = xs[k0 + 2 * half][n0 + l16];        // B[tok,c]
      bf.y = xs[k0 + 2 * half + 1][n0 + l16];
      acc = __builtin_amdgcn_wmma_f32_16x16x4_f32(false, a, false, bf,
                                                  (short)0, acc, false, false);
    }
    if (half == 0) {                             // rows 8..15 are slot padding
#pragma unroll
      for (int g = 0; g < 8; ++g)
        atomicAdd(&raw_ws[(b * NS_ + g) * D_ + n0 + l16], acc[g]);
    }
  }
  // per-slot attention-mass column sums (8 columns, conflict-free stride 17)
  if (t < NS_) {
    float s = 0.f;
    for (int r = 0; r < TILE_; ++r) s += at[r][t];
    atomicAdd(&asum_ws[b * NS_ + t], s);
  }
}

// ---------------------------------------------------------------------------
// Per-batch slot update: normalize, apply Wv^T + bv, GRU cell, residual MLP.
// All [8 x 64]-sized work: plain VALU, one block per batch.
// ---------------------------------------------------------------------------
__global__ void __launch_bounds__(256)
slot_update_kernel(const float* __restrict__ Wv, const float* __restrict__ bv,
                   const float* __restrict__ W_ih, const float* __restrict__ b_ih,
                   const float* __restrict__ W_hh, const float* __restrict__ b_hh,
                   const float* __restrict__ W1, const float* __restrict__ b1,
                   const float* __restrict__ W2, const float* __restrict__ b2,
                   const float* __restrict__ ln_ff_w, const float* __restrict__ ln_ff_b,
                   const float* __restrict__ raw_ws, const float* __restrict__ asum_ws,
                   float* __restrict__ slots_ws, float* __restrict__ out, int last) {
  const int b = blockIdx.x, t = threadIdx.x;
  __shared__ float u1n[NS_][D_];     // raw / attn_sum
  __shared__ float upd[NS_][D_];     // updates after Wv^T + bv
  __shared__ float sprev[NS_][D_];
  __shared__ float gx[NS_][3 * D_], gh[NS_][3 * D_];
  __shared__ float smid[NS_][D_];
  __shared__ float ff[NS_][D_];
  __shared__ float h1[NS_][H_];
  __shared__ float inv_s[NS_], mu_s[NS_], rs_s[NS_];

  if (t < NS_) inv_s[t] = 1.f / asum_ws[b * NS_ + t];
  for (int idx = t; idx < NS_ * D_; idx += 256)
    sprev[idx >> 6][idx & 63] = slots_ws[b * NS_ * D_ + idx];
  __syncthreads();
  for (int idx = t; idx < NS_ * D_; idx += 256)
    u1n[idx >> 6][idx & 63] = raw_ws[b * NS_ * D_ + idx] * inv_s[idx >> 6];
  __syncthreads();
  for (int idx = t; idx < NS_ * D_; idx += 256) {
    const int j = idx >> 6, c = idx & 63;
    float acc = bv[c];
    for (int cc = 0; cc < D_; ++cc) acc += u1n[j][cc] * Wv[c * D_ + cc];
    upd[j][c] = acc;
  }
  __syncthreads();
  // GRU gates
  for (int idx = t; idx < NS_ * 3 * D_; idx += 256) {
    const int j = idx / (3 * D_), r = idx % (3 * D_);
    float ax = b_ih[r], ah = b_hh[r];
    for (int c = 0; c < D_; ++c) {
      ax += upd[j][c] * W_ih[r * D_ + c];
      ah += sprev[j][c] * W_hh[r * D_ + c];
    }
    gx[j][r] = ax; gh[j][r] = ah;
  }
  __syncthreads();
  for (int idx = t; idx < NS_ * D_; idx += 256) {
    const int j = idx >> 6, c = idx & 63;
    const float r = 1.f / (1.f + __expf(-(gx[j][c] + gh[j][c])));
    const float z = 1.f / (1.f + __expf(-(gx[j][D_ + c] + gh[j][D_ + c])));
    const float nn = tanhf(gx[j][2 * D_ + c] + r * gh[j][2 * D_ + c]);
    smid[j][c] = (1.f - z) * nn + z * sprev[j][c];
  }
  __syncthreads();
  if (t < NS_) {
    float m = 0.f;
    for (int c = 0; c < D_; ++c) m += smid[t][c];
    m *= (1.f / D_);
    float v = 0.f;
    for (int c = 0; c < D_; ++c) { float d = smid[t][c] - m; v += d * d; }
    v *= (1.f / D_);
    mu_s[t] = m; rs_s[t] = rsqrtf(v + LN_EPS_);
  }
  __syncthreads();
  for (int idx = t; idx < NS_ * D_; idx += 256) {
    const int j = idx >> 6, c = idx & 63;
    ff[j][c] = (smid[j][c] - mu_s[j]) * rs_s[j] * ln_ff_w[c] + ln_ff_b[c];
  }
  __syncthreads();
  for (int idx = t; idx < NS_ * H_; idx += 256) {
    const int j = idx >> 7, r = idx & 127;
    float acc = b1[r];
    for (int c = 0; c < D_; ++c) acc += ff[j][c] * W1[r * D_ + c];
    h1[j][r] = fmaxf(acc, 0.f);
  }
  __syncthreads();
  for (int idx = t; idx < NS_ * D_; idx += 256) {
    const int j = idx >> 6, c = idx & 63;
    float acc = b2[c];
    for (int r = 0; r < H_; ++r) acc += h1[j][r] * W2[c * H_ + r];
    const float sv = smid[j][c] + acc;
    slots_ws[b * NS_ * D_ + idx] = sv;
    if (last) out[b * NS_ * D_ + idx] = sv;
  }
}

// ---------------------------------------------------------------------------
extern "C" void kernel_launch(void* const* d_in, const int* in_sizes, int n_in,
                              void* d_out, int out_size, void* d_ws, size_t ws_size,
                              hipStream_t stream) {
  const float* inputs      = (const float*)d_in[0];
  const int*   mask        = (const int*)  d_in[1];
  const float* noise       = (const float*)d_in[2];
  const float* slots_mu    = (const float*)d_in[3];
  const float* slots_sigma = (const float*)d_in[4];
  const float* Wq = (const float*)d_in[5];  const float* bq = (const float*)d_in[6];
  const float* Wk = (const float*)d_in[7];  const float* bk = (const float*)d_in[8];
  const float* Wv = (const float*)d_in[9];  const float* bv = (const float*)d_in[10];
  const float* W_ih = (const float*)d_in[11]; const float* b_ih = (const float*)d_in[12];
  const float* W_hh = (const float*)d_in[13]; const float* b_hh = (const float*)d_in[14];
  const float* W1 = (const float*)d_in[15]; const float* b1 = (const float*)d_in[16];
  const float* W2 = (const float*)d_in[17]; const float* b2 = (const float*)d_in[18];
  const float* ln_in_w = (const float*)d_in[19]; const float* ln_in_b = (const float*)d_in[20];
  const float* ln_s_w  = (const float*)d_in[21]; const float* ln_s_b  = (const float*)d_in[22];
  const float* ln_ff_w = (const float*)d_in[23]; const float* ln_ff_b = (const float*)d_in[24];
  float* out = (float*)d_out;

  // workspace layout (floats): ~518KB total
  float* ws       = (float*)d_ws;
  float* slots_ws = ws;                              // B*8*64
  float* qk_ws    = slots_ws + B_ * NS_ * D_;        // B*16*64
  float* qb_ws    = qk_ws + B_ * NSP_ * D_;          // B*16
  float* raw_ws   = qb_ws + B_ * NSP_;               // B*8*64
  float* asum_ws  = raw_ws + B_ * NS_ * D_;          // B*8

  for (int it = 0; it < 3; ++it) {
    slot_q_kernel<<<B_, 128, 0, stream>>>(
        noise, slots_mu, slots_sigma, Wq, bq, Wk, bk, ln_s_w, ln_s_b,
        slots_ws, qk_ws, qb_ws, raw_ws, asum_ws, it);
    slot_attn_kernel<<<dim3(B_, N_ / TILE_), 128, 0, stream>>>(
        inputs, mask, ln_in_w, ln_in_b, qk_ws, qb_ws, raw_ws, asum_ws);
    slot_update_kernel<<<B_, 256, 0, stream>>>(
        Wv, bv, W_ih, b_ih, W_hh, b_hh, W1, b1, W2, b2, ln_ff_w, ln_ff_b,
        raw_ws, asum_ws, slots_ws, out, it == 2 ? 1 : 0);
  }
}